// MambaBlock_52458730553553
// MI455X (gfx1250) — compile-verified
//
#include <hip/hip_runtime.h>
#include <cmath>

#define DIMQ    512
#define DSTATE  16
#define DINNER  1024
#define BATCH   4
#define SEQLEN  4096
#define NTOK    (BATCH * SEQLEN)   // 16384

typedef __attribute__((ext_vector_type(2))) float v2f;
typedef __attribute__((ext_vector_type(8))) float v8f;
typedef __attribute__((__vector_size__(4 * sizeof(int)))) int v4i_t;

// Pad LDS tile rows to 20 floats (80B): every 16B chunk stays 16B-aligned and
// stride-20 is bank-conflict-free across 16 lanes (gcd(20,64)=4 -> 16 banks).
#define LPAD 20

// ---- async global->LDS staging (gfx1250 GLOBAL_LOAD_ASYNC_TO_LDS_B128) ----
#if __has_builtin(__builtin_amdgcn_global_load_async_to_lds_b128) && \
    __has_builtin(__builtin_amdgcn_s_wait_asynccnt)
#define USE_ASYNC_LDS 1
#else
#define USE_ASYNC_LDS 0
#endif

__device__ __forceinline__ void g2l_b128(const float* g, float* l) {
#if USE_ASYNC_LDS
    __builtin_amdgcn_global_load_async_to_lds_b128(
        (__attribute__((address_space(1))) v4i_t*)(g),
        (__attribute__((address_space(3))) v4i_t*)(l), 0, 0);
#else
    *(float4*)l = *(const float4*)g;
#endif
}

__device__ __forceinline__ void async_stage_wait() {
#if USE_ASYNC_LDS
    __builtin_amdgcn_s_wait_asynccnt(0);
#endif
}

__device__ __forceinline__ float silu_f(float v) {
    return v / (1.0f + expf(-v));
}

// ---------------------------------------------------------------------------
// GEMM 1: out[m][n] = sum_k X[m][k] * W[n][k]
// X: (16384, 512), W: (2048, 512), out: (16384, 2048)  (xi | res fused buffer)
// Block: 256 threads (8 waves). Tile: 128(M) x 128(N), BK = 16.
// Wave w owns rows [w*16, w*16+16) and all 8 N-subtiles (v8f acc each).
// ---------------------------------------------------------------------------
__global__ __launch_bounds__(256) void gemm_in_kernel(const float* __restrict__ X,
                                                      const float* __restrict__ W,
                                                      float* __restrict__ out) {
    __shared__ __align__(16) float As[128 * LPAD];   // As[m][k]
    __shared__ __align__(16) float Bs[128 * LPAD];   // Bs[n][k]

    const int m0   = blockIdx.y * 128;
    const int n0   = blockIdx.x * 128;
    const int tid  = threadIdx.x;
    const int lane = tid & 31;
    const int wave = tid >> 5;
    const int lp   = lane & 15;
    const int hi   = lane >> 4;
    const int wm   = wave * 16 + lp;

    v8f acc[8] = {};

    const int r  = tid >> 1;          // 0..127 : tile row
    const int c8 = (tid & 1) * 8;     // 0 or 8 : k-offset

    for (int k0 = 0; k0 < 512; k0 += 16) {
        // Stage A tile (128 rows x 16 k) -> As[m][k], 2 x B128 per thread
        const float* ag = X + (size_t)(m0 + r) * 512 + k0 + c8;
        g2l_b128(ag,     &As[r * LPAD + c8]);
        g2l_b128(ag + 4, &As[r * LPAD + c8 + 4]);
        // Stage B tile: Bs[n][k] = W[(n0+n)*512 + k0 + k], 2 x B128 per thread
#pragma unroll
        for (int i = 0; i < 2; ++i) {
            int flat = tid + 256 * i;   // 0..511
            int nn   = flat >> 2;       // 0..127
            int kq   = (flat & 3) * 4;  // 0,4,8,12
            g2l_b128(W + (size_t)(n0 + nn) * 512 + k0 + kq, &Bs[nn * LPAD + kq]);
        }
        if (k0 + 16 < 512) __builtin_prefetch(ag + 16, 0, 1);
        async_stage_wait();
        __syncthreads();

#pragma unroll
        for (int ks = 0; ks < 16; ks += 4) {
            // A frag: a[j] = A[m][ks + j + 2*hi]  (ISA 16x4 f32 layout)
            v2f a;
            a[0] = As[wm * LPAD + ks + 2 * hi + 0];
            a[1] = As[wm * LPAD + ks + 2 * hi + 1];
#pragma unroll
            for (int nt = 0; nt < 8; ++nt) {
                v2f b;
                b[0] = Bs[(nt * 16 + lp) * LPAD + ks + 2 * hi + 0];
                b[1] = Bs[(nt * 16 + lp) * LPAD + ks + 2 * hi + 1];
                acc[nt] = __builtin_amdgcn_wmma_f32_16x16x4_f32(
                    false, a, false, b, (short)0, acc[nt], false, false);
            }
        }
        __syncthreads();
    }

    // D layout: row = rr + 8*hi, col = lp within 16x16 tile
#pragma unroll
    for (int nt = 0; nt < 8; ++nt) {
#pragma unroll
        for (int rr = 0; rr < 8; ++rr) {
            int row = m0 + wave * 16 + rr + 8 * hi;
            int col = n0 + nt * 16 + lp;
            out[(size_t)row * 2048 + col] = acc[nt][rr];
        }
    }
}

// ---------------------------------------------------------------------------
// Depthwise causal conv (K=4, left pad 3) + bias + SiLU.
// xi lives in xr[:, :1024] (token stride 2048). xa: (16384, 1024).
// ---------------------------------------------------------------------------
__global__ __launch_bounds__(256) void conv_silu_kernel(const float* __restrict__ xr,
                                                        const float* __restrict__ cw,
                                                        const float* __restrict__ cb,
                                                        float* __restrict__ xa) {
    const int bx = blockIdx.x;
    const int cg = bx & 3;            // channel group (4 x 256)
    const int tc = (bx >> 2) & 255;   // token chunk (256 x 16)
    const int b  = bx >> 10;          // batch
    const int c  = cg * 256 + threadIdx.x;
    const int t0 = tc * 16;

    const float w0 = cw[c * 4 + 0], w1 = cw[c * 4 + 1];
    const float w2 = cw[c * 4 + 2], w3 = cw[c * 4 + 3];
    const float bias = cb[c];

    const float* xi = xr + (size_t)b * SEQLEN * 2048 + c;   // token stride 2048
    float h0 = (t0 >= 3) ? xi[(size_t)(t0 - 3) * 2048] : 0.0f;
    float h1 = (t0 >= 2) ? xi[(size_t)(t0 - 2) * 2048] : 0.0f;
    float h2 = (t0 >= 1) ? xi[(size_t)(t0 - 1) * 2048] : 0.0f;
    float* xo = xa + ((size_t)b * SEQLEN + t0) * 1024 + c;

#pragma unroll
    for (int t = 0; t < 16; ++t) {
        float cur = xi[(size_t)(t0 + t) * 2048];
        float v = w0 * h0 + w1 * h1 + w2 * h2 + w3 * cur + bias;
        xo[(size_t)t * 1024] = silu_f(v);
        h0 = h1; h1 = h2; h2 = cur;
    }
}

// ---------------------------------------------------------------------------
// Per-token: dt = softplus(W_dt@xa + b_dt), B_proj = (W_x@xa)[16:32],
// u = mean(xa[:64]); emits A_disc = exp(-exp(A_log)*dt) and Bu = dt*B_proj*u.
// One wave32 per token; butterfly shuffle reduction.
// ---------------------------------------------------------------------------
__global__ __launch_bounds__(256) void proj_kernel(const float* __restrict__ xa,
                                                   const float* __restrict__ Wx,
                                                   const float* __restrict__ Wdt,
                                                   const float* __restrict__ bdt,
                                                   const float* __restrict__ Alog,
                                                   float* __restrict__ Ad,
                                                   float* __restrict__ Bu) {
    const int wave  = threadIdx.x >> 5;
    const int lane  = threadIdx.x & 31;
    const int token = blockIdx.x * 8 + wave;
    const float* row = xa + (size_t)token * 1024;

    float dt_acc[16], bp_acc[16];
#pragma unroll
    for (int q = 0; q < 16; ++q) { dt_acc[q] = 0.0f; bp_acc[q] = 0.0f; }
    float u_acc = 0.0f;

    for (int c = lane; c < 1024; c += 32) {
        float v = row[c];
        if (c < 64) u_acc += v;
#pragma unroll
        for (int q = 0; q < 16; ++q) {
            dt_acc[q] += Wdt[q * 1024 + c] * v;
            bp_acc[q] += Wx[(16 + q) * 1024 + c] * v;
        }
    }

#pragma unroll
    for (int off = 16; off > 0; off >>= 1) {
        u_acc += __shfl_xor(u_acc, off, 32);
#pragma unroll
        for (int q = 0; q < 16; ++q) {
            dt_acc[q] += __shfl_xor(dt_acc[q], off, 32);
            bp_acc[q] += __shfl_xor(bp_acc[q], off, 32);
        }
    }

    const float u = u_acc * (1.0f / 64.0f);
    float dtv = 0.0f, bpv = 0.0f;
#pragma unroll
    for (int q = 0; q < 16; ++q) {
        if (lane == q) { dtv = dt_acc[q]; bpv = bp_acc[q]; }
    }

    if (lane < DSTATE) {
        float pre = dtv + bdt[lane];
        float dt  = (pre > 20.0f) ? pre : log1pf(expf(pre));
        float An  = -expf(Alog[lane]);
        Ad[(size_t)token * 16 + lane] = expf(An * dt);
        Bu[(size_t)token * 16 + lane] = dt * bpv * u;
    }
}

// ---------------------------------------------------------------------------
// Linear recurrence s <- s*a + b over L=4096, one block per (batch, state).
// Chunk-fold (P,Q) per thread + Hillis-Steele scan in LDS + replay.
// ---------------------------------------------------------------------------
__global__ __launch_bounds__(256) void scan_kernel(const float* __restrict__ Ad,
                                                   const float* __restrict__ Bu,
                                                   float* __restrict__ ys) {
    __shared__ float sa[256];
    __shared__ float sb[256];
    const int b = blockIdx.x >> 4;
    const int n = blockIdx.x & 15;
    const int i = threadIdx.x;
    const size_t base = ((size_t)b * SEQLEN + (size_t)i * 16) * 16 + n;

    float P = 1.0f, Q = 0.0f;
#pragma unroll
    for (int t = 0; t < 16; ++t) {
        float a  = Ad[base + (size_t)t * 16];
        float bu = Bu[base + (size_t)t * 16];
        P *= a;
        Q = Q * a + bu;
    }
    sa[i] = P; sb[i] = Q;
    __syncthreads();

    for (int off = 1; off < 256; off <<= 1) {
        float pa = 1.0f, pb = 0.0f;
        if (i >= off) { pa = sa[i - off]; pb = sb[i - off]; }
        __syncthreads();
        if (i >= off) {
            sb[i] = sb[i] + sa[i] * pb;   // uses old sa[i]
            sa[i] = sa[i] * pa;
        }
        __syncthreads();
    }

    float s = (i == 0) ? 0.0f : sb[i - 1];   // s0 = 0 -> carry is Q-prefix
#pragma unroll
    for (int t = 0; t < 16; ++t) {
        float a  = Ad[base + (size_t)t * 16];
        float bu = Bu[base + (size_t)t * 16];
        s = s * a + bu;
        ys[base + (size_t)t * 16] = s;
    }
}

// ---------------------------------------------------------------------------
// GEMM 2: out[m][n] = sum_k A[m][k] * W_out[n][k], with
// A[m][k] = (ys[m][k/64] + xa[m][k]*D[k]) * silu(res[m][k]) fused into staging.
// A: (16384, 1024), W_out: (512, 1024), out: (16384, 512).
// ---------------------------------------------------------------------------
__global__ __launch_bounds__(256) void gemm_out_kernel(const float* __restrict__ xa,
                                                       const float* __restrict__ xr,
                                                       const float* __restrict__ ys,
                                                       const float* __restrict__ Dp,
                                                       const float* __restrict__ W,
                                                       float* __restrict__ out) {
    __shared__ __align__(16) float As[128 * LPAD];   // As[m][k]
    __shared__ __align__(16) float Bs[128 * LPAD];   // Bs[n][k]

    const int m0   = blockIdx.y * 128;
    const int n0   = blockIdx.x * 128;
    const int tid  = threadIdx.x;
    const int lane = tid & 31;
    const int wave = tid >> 5;
    const int lp   = lane & 15;
    const int hi   = lane >> 4;
    const int wm   = wave * 16 + lp;

    v8f acc[8] = {};

    const int r   = tid >> 1;
    const int c8  = (tid & 1) * 8;
    const int tok = m0 + r;

    for (int k0 = 0; k0 < 1024; k0 += 16) {
        // Stage fused A tile -> As[m][k] (computed elementwise, ds stores)
#pragma unroll
        for (int ii = 0; ii < 8; ++ii) {
            int c = k0 + c8 + ii;
            float resv = xr[(size_t)tok * 2048 + 1024 + c];
            float yv = ys[(size_t)tok * 16 + (c >> 6)] + xa[(size_t)tok * 1024 + c] * Dp[c];
            As[r * LPAD + c8 + ii] = yv * silu_f(resv);
        }
        // Stage B tile: Bs[n][k] = W[(n0+n)*1024 + k0 + k], 2 x B128 per thread
#pragma unroll
        for (int i = 0; i < 2; ++i) {
            int flat = tid + 256 * i;
            int nn   = flat >> 2;
            int kq   = (flat & 3) * 4;
            g2l_b128(W + (size_t)(n0 + nn) * 1024 + k0 + kq, &Bs[nn * LPAD + kq]);
        }
        if (k0 + 16 < 1024) __builtin_prefetch(xa + (size_t)tok * 1024 + k0 + 16 + c8, 0, 1);
        async_stage_wait();
        __syncthreads();

#pragma unroll
        for (int ks = 0; ks < 16; ks += 4) {
            v2f a;
            a[0] = As[wm * LPAD + ks + 2 * hi + 0];
            a[1] = As[wm * LPAD + ks + 2 * hi + 1];
#pragma unroll
            for (int nt = 0; nt < 8; ++nt) {
                v2f b;
                b[0] = Bs[(nt * 16 + lp) * LPAD + ks + 2 * hi + 0];
                b[1] = Bs[(nt * 16 + lp) * LPAD + ks + 2 * hi + 1];
                acc[nt] = __builtin_amdgcn_wmma_f32_16x16x4_f32(
                    false, a, false, b, (short)0, acc[nt], false, false);
            }
        }
        __syncthreads();
    }

#pragma unroll
    for (int nt = 0; nt < 8; ++nt) {
#pragma unroll
        for (int rr = 0; rr < 8; ++rr) {
            int row = m0 + wave * 16 + rr + 8 * hi;
            int col = n0 + nt * 16 + lp;
            out[(size_t)row * 512 + col] = acc[nt][rr];
        }
    }
}

// ---------------------------------------------------------------------------
extern "C" void kernel_launch(void* const* d_in, const int* in_sizes, int n_in,
                              void* d_out, int out_size, void* d_ws, size_t ws_size,
                              hipStream_t stream) {
    (void)in_sizes; (void)n_in; (void)out_size; (void)ws_size;

    const float* x     = (const float*)d_in[0];  // (4, 4096, 512)
    const float* W_in  = (const float*)d_in[1];  // (2048, 512)
    const float* cw    = (const float*)d_in[2];  // (1024, 1, 4)
    const float* cb    = (const float*)d_in[3];  // (1024,)
    const float* W_x   = (const float*)d_in[4];  // (32, 1024)
    const float* W_dt  = (const float*)d_in[5];  // (16, 1024)
    const float* b_dt  = (const float*)d_in[6];  // (16,)
    const float* A_log = (const float*)d_in[7];  // (1, 16)
    const float* Dp    = (const float*)d_in[8];  // (1024,)
    const float* W_out = (const float*)d_in[9];  // (512, 1024)
    float* out = (float*)d_out;                  // (4, 4096, 512)

    // Workspace layout (floats)
    float* xr = (float*)d_ws;                    // (16384, 2048)  xi|res
    float* xa = xr + (size_t)NTOK * 2048;        // (16384, 1024)
    float* Ad = xa + (size_t)NTOK * 1024;        // (16384, 16)
    float* Bu = Ad + (size_t)NTOK * 16;          // (16384, 16)
    float* ys = Bu + (size_t)NTOK * 16;          // (16384, 16)

    gemm_in_kernel  <<<dim3(2048 / 128, NTOK / 128), 256, 0, stream>>>(x, W_in, xr);
    conv_silu_kernel<<<BATCH * (SEQLEN / 16) * (DINNER / 256), 256, 0, stream>>>(xr, cw, cb, xa);
    proj_kernel     <<<NTOK / 8, 256, 0, stream>>>(xa, W_x, W_dt, b_dt, A_log, Ad, Bu);
    scan_kernel     <<<BATCH * DSTATE, 256, 0, stream>>>(Ad, Bu, ys);
    gemm_out_kernel <<<dim3(DIMQ / 128, NTOK / 128), 256, 0, stream>>>(xa, xr, ys, Dp, W_out, out);
}